// Q_Liner_60129542144045
// MI455X (gfx1250) — compile-verified
//
#include <hip/hip_runtime.h>
#include <hip/hip_bf16.h>

// ---------------------------------------------------------------------------
// 12-qubit statevector simulator, batch=1024, DIM=4096 complex amplitudes.
// One workgroup (256 threads = 8 wave32) per batch element; the state lives
// entirely in LDS (32 KB) for all 6 rotation layers + 5 CNOT rings.
//
// Per layer: 12 single-qubit gates fused into three 16x16 complex unitaries
// (Kronecker product over 4-qubit nibbles). Applying one 16x16 gate =
// GEMM U(16x16) x S(16x256): 16 column blocks of 16x16x16 via chained
// V_WMMA_F32_16X16X4_F32 (fp32 matrix path keeps complex64 precision).
// Complex GEMM = 4 real GEMMs with accumulator chaining.
// CNOT ring = composed index permutation, one LDS gather/scatter.
//
// v2 changes: group index is a template parameter (no runtime branches in
// LDS address math), per-qubit 2x2 gates computed once per layer by 12
// threads (kills the sincos storm), single gate-build phase per layer.
// ---------------------------------------------------------------------------

#define DIM       4096
#define N_QUBITS  12
#define N_LAYERS  5

typedef __attribute__((ext_vector_type(2))) float v2f;
typedef __attribute__((ext_vector_type(8))) float v8f;

struct cx { float r, i; };
__device__ __forceinline__ cx cmul(cx a, cx b) { return { a.r*b.r - a.i*b.i, a.r*b.i + a.i*b.r }; }
__device__ __forceinline__ cx cadd(cx a, cx b) { return { a.r + b.r, a.i + b.i }; }

struct m2 { cx e[2][2]; };

__device__ __forceinline__ m2 mmul(const m2& A, const m2& B) {
  m2 C;
#pragma unroll
  for (int i = 0; i < 2; ++i)
#pragma unroll
    for (int j = 0; j < 2; ++j)
      C.e[i][j] = cadd(cmul(A.e[i][0], B.e[0][j]), cmul(A.e[i][1], B.e[1][j]));
  return C;
}

__device__ __forceinline__ m2 rot_x(float t) {
  float c = cosf(0.5f * t), s = sinf(0.5f * t);
  m2 M; M.e[0][0] = {c, 0}; M.e[0][1] = {0, -s};
        M.e[1][0] = {0, -s}; M.e[1][1] = {c, 0};
  return M;
}
__device__ __forceinline__ m2 rot_y(float t) {
  float c = cosf(0.5f * t), s = sinf(0.5f * t);
  m2 M; M.e[0][0] = {c, 0}; M.e[0][1] = {-s, 0};
        M.e[1][0] = {s, 0}; M.e[1][1] = {c, 0};
  return M;
}
__device__ __forceinline__ m2 rot_z(float t) {
  float c = cosf(0.5f * t), s = sinf(0.5f * t);
  m2 M; M.e[0][0] = {c, -s}; M.e[0][1] = {0, 0};
        M.e[1][0] = {0, 0};  M.e[1][1] = {c, s};
  return M;
}

// Qubit q acts on state-index bit (11 - q) (row-major reshape in reference).
// Group QB covers one nibble: QB=0 -> bits 11..8, QB=4 -> 7..4, QB=8 -> 3..0.
// r = index within the nibble, c = flattened index of the other 8 bits.
template <int QB>
__device__ __forceinline__ constexpr int sidx(int r, int c) {
  if constexpr (QB == 0) return (r << 8) | c;                           // hi
  else if constexpr (QB == 4) return ((c & 0xF0) << 4) | (r << 4) | (c & 15); // mid
  else return (c << 4) | r;                                             // lo
}

__device__ __forceinline__ v8f wmma_f32(v2f a, v2f b, v8f c) {
  // V_WMMA_F32_16X16X4_F32: D(16x16,f32) = A(16x4,f32) x B(4x16,f32) + C
  return __builtin_amdgcn_wmma_f32_16x16x4_f32(
      false, a, false, b, (short)0, c, false, false);
}

// Apply the 16x16 complex gate (ure/uim, row-major) to the LDS state on
// nibble QB. Each wave owns 2 of 16 column blocks. Per block:
//   Dre = Ure*Sre - Uim*Sim ; Dim = Ure*Sim + Uim*Sre
// each real product = 4 chained 16x16x4 WMMAs (K = 16 in chunks of 4).
template <int QB>
__device__ void apply_group(float* sre, float* sim,
                            const float* ure, const float* uim) {
  const int lane  = threadIdx.x & 31;
  const int wave  = threadIdx.x >> 5;
  const int half  = lane >> 4;      // 0: lanes 0-15, 1: lanes 16-31
  const int n     = lane & 15;      // N index (and A's M index)
  const int khalf = half << 1;      // K offset for the upper half-wave

  // A fragments (the gate) — invariant across column blocks.
  // ISA 32-bit A 16x4 layout: VGPR0 = K{0,2}, VGPR1 = K{1,3}.
  v2f aR[4], aI[4], aIn[4];
#pragma unroll
  for (int kc = 0; kc < 4; ++kc) {
    const int k0 = kc * 4 + khalf;              // even -> 8B aligned
    aR[kc] = *(const v2f*)&ure[n * 16 + k0];
    aI[kc] = *(const v2f*)&uim[n * 16 + k0];
    aIn[kc] = -aI[kc];
  }

#pragma unroll
  for (int blk = 0; blk < 2; ++blk) {
    const int cb = wave * 2 + blk;              // column block 0..15
    const int c  = cb * 16 + n;

    // B fragments (state slice), mirrored layout: VGPR0 = K{0,2}, VGPR1 = K{1,3}.
    v2f bR[4], bI[4];
#pragma unroll
    for (int kc = 0; kc < 4; ++kc) {
      const int k0 = kc * 4 + khalf;
      if constexpr (QB == 8) {
        // lo nibble: consecutive K are contiguous in LDS -> b64 loads
        bR[kc] = *(const v2f*)&sre[(c << 4) + k0];
        bI[kc] = *(const v2f*)&sim[(c << 4) + k0];
      } else {
        bR[kc].x = sre[sidx<QB>(k0,     c)];
        bR[kc].y = sre[sidx<QB>(k0 + 1, c)];
        bI[kc].x = sim[sidx<QB>(k0,     c)];
        bI[kc].y = sim[sidx<QB>(k0 + 1, c)];
      }
    }

    v8f dR = {};
    v8f dI = {};
#pragma unroll
    for (int kc = 0; kc < 4; ++kc) dR = wmma_f32(aR[kc],  bR[kc], dR);
#pragma unroll
    for (int kc = 0; kc < 4; ++kc) dR = wmma_f32(aIn[kc], bI[kc], dR);
#pragma unroll
    for (int kc = 0; kc < 4; ++kc) dI = wmma_f32(aR[kc],  bI[kc], dI);
#pragma unroll
    for (int kc = 0; kc < 4; ++kc) dI = wmma_f32(aI[kc],  bR[kc], dI);

    // D layout: VGPR v -> M = v + 8*half, N = lane&15.
    if constexpr (QB == 8) {
      const int base = (c << 4) + 8 * half;     // consecutive v contiguous
#pragma unroll
      for (int v = 0; v < 4; ++v) {
        v2f pr = { dR[2 * v], dR[2 * v + 1] };
        v2f pi = { dI[2 * v], dI[2 * v + 1] };
        *(v2f*)&sre[base + 2 * v] = pr;
        *(v2f*)&sim[base + 2 * v] = pi;
      }
    } else {
#pragma unroll
      for (int v = 0; v < 8; ++v) {
        const int m   = v + 8 * half;
        const int idx = sidx<QB>(m, c);
        sre[idx] = dR[v];
        sim[idx] = dI[v];
      }
    }
  }
}

// Ring of 12 CNOTs composed into one index permutation.
// CNOT(q,(q+1)%12): control bit 11-q, target bit 11-((q+1)%12), each an
// involution sigma_q(j) = j ^ (bit_c(j) << t). new[i] = old[s0(s1(...s11(i)))].
__device__ void ring_cnot(float* sre, float* sim) {
  const int tid = threadIdx.x;
  float tr[16], ti[16];
#pragma unroll
  for (int v = 0; v < 16; ++v) {
    const int i = tid + (v << 8);
    int j = i;
#pragma unroll
    for (int q = 11; q >= 0; --q) {
      const int cb = 11 - q;
      const int tb = 11 - ((q + 1) % 12);
      j ^= ((j >> cb) & 1) << tb;
    }
    tr[v] = sre[j];
    ti[v] = sim[j];
  }
  __syncthreads();
#pragma unroll
  for (int v = 0; v < 16; ++v) {
    const int i = tid + (v << 8);
    sre[i] = tr[v];
    sim[i] = ti[v];
  }
  __syncthreads();
}

__global__ __launch_bounds__(256)
void qsim_kernel(const float* __restrict__ x, const float* __restrict__ w,
                 float* __restrict__ out) {
  __shared__ float sre[DIM];
  __shared__ float sim[DIM];
  __shared__ float ure[3][256];       // the three 16x16 gates of this layer
  __shared__ float uim[3][256];
  __shared__ float g2re[N_QUBITS][4]; // per-qubit fused 2x2 (e[mb][jb] -> mb*2+jb)
  __shared__ float g2im[N_QUBITS][4];
  __shared__ float red[256];

  const int b   = blockIdx.x;
  const int tid = threadIdx.x;
  const float* xb = x + (size_t)b * DIM;

  // Amplitude encoding: psi = x / sqrt(||x|| + 1e-12)  (faithful to reference)
  float xs[16];
  float ss = 0.f;
#pragma unroll
  for (int v = 0; v < 16; ++v) {
    const float t = xb[tid + (v << 8)];
    xs[v] = t;
    ss += t * t;
  }
  red[tid] = ss;
  __syncthreads();
  for (int s = 128; s > 0; s >>= 1) {
    if (tid < s) red[tid] += red[tid + s];
    __syncthreads();
  }
  const float scale = rsqrtf(sqrtf(red[0]) + 1e-12f);
#pragma unroll
  for (int v = 0; v < 16; ++v) {
    const int i = tid + (v << 8);
    sre[i] = xs[v] * scale;
    sim[i] = 0.f;
  }
  __syncthreads();

  for (int l = 0; l <= N_LAYERS; ++l) {
    // --- Phase 1: 12 threads compute the 12 fused per-qubit 2x2 gates -----
    if (tid < N_QUBITS) {
      m2 U;
      if (l < N_LAYERS) {
        const float* p = w + l * 36 + tid * 3;              // XYZ: Rz*Ry*Rx
        U = mmul(rot_z(p[2]), mmul(rot_y(p[1]), rot_x(p[0])));
      } else {
        const float* p = w + N_LAYERS * 36 + tid * 3;       // YZY: Ry*Rz*Ry
        U = mmul(rot_y(p[2]), mmul(rot_z(p[1]), rot_y(p[0])));
      }
#pragma unroll
      for (int e = 0; e < 4; ++e) {
        g2re[tid][e] = U.e[e >> 1][e & 1].r;
        g2im[tid][e] = U.e[e >> 1][e & 1].i;
      }
    }
    __syncthreads();

    // --- Phase 2: build all three 16x16 Kronecker gates (1 entry x 3/thread)
    {
      const int m = tid >> 4, j = tid & 15;
#pragma unroll
      for (int g = 0; g < 3; ++g) {
        cx prod = {1.f, 0.f};
#pragma unroll
        for (int k = 0; k < 4; ++k) {
          const int q  = g * 4 + k;
          const int mb = (m >> (3 - k)) & 1;
          const int jb = (j >> (3 - k)) & 1;
          const cx u = { g2re[q][mb * 2 + jb], g2im[q][mb * 2 + jb] };
          prod = cmul(prod, u);
        }
        ure[g][tid] = prod.r;
        uim[g][tid] = prod.i;
      }
    }
    __syncthreads();

    // --- Phase 3: apply the three group gates with WMMA ------------------
    apply_group<0>(sre, sim, ure[0], uim[0]);
    __syncthreads();
    apply_group<4>(sre, sim, ure[1], uim[1]);
    __syncthreads();
    apply_group<8>(sre, sim, ure[2], uim[2]);
    __syncthreads();

    // --- Phase 4: ring of CNOTs (not after the final YZY layer) ----------
    if (l < N_LAYERS) ring_cnot(sre, sim);
  }

  float* ob = out + (size_t)b * DIM;
#pragma unroll
  for (int v = 0; v < 16; ++v) {
    const int i = tid + (v << 8);
    ob[i] = sre[i];
  }
}

extern "C" void kernel_launch(void* const* d_in, const int* in_sizes, int n_in,
                              void* d_out, int out_size, void* d_ws, size_t ws_size,
                              hipStream_t stream) {
  const float* x = (const float*)d_in[0];   // [1024, 4096] f32
  const float* w = (const float*)d_in[1];   // [216] f32
  float* out = (float*)d_out;               // [1024, 4096] f32
  const int batch = in_sizes[0] >> 12;      // DIM = 4096
  qsim_kernel<<<dim3(batch), dim3(256), 0, stream>>>(x, w, out);
}